// GCN_28037546508929
// MI455X (gfx1250) — compile-verified
//
#include <hip/hip_runtime.h>
#include <hip/hip_bf16.h>

typedef __attribute__((ext_vector_type(2))) float v2f;
typedef __attribute__((ext_vector_type(8))) float v8f;

#define NEG_SLOPE 0.1f

// ---------------- degree / normalization ----------------
__global__ void k_deg_init(float* deg, int n) {
    int i = blockIdx.x * blockDim.x + threadIdx.x;
    if (i < n) deg[i] = 1.0f;   // self-loop contributes 1 to every dst degree
}

__global__ void k_deg_accum(const int* __restrict__ dst, float* deg, int E) {
    int e = blockIdx.x * blockDim.x + threadIdx.x;
    if (e < E) atomicAdd(&deg[dst[e]], 1.0f);
}

__global__ void k_deg_rsqrt(float* deg, int n) {
    int i = blockIdx.x * blockDim.x + threadIdx.x;
    if (i < n) deg[i] = rsqrtf(deg[i]);   // deg >= 1 always
}

// ---------------- weight transpose: Wt[n*K+k] = W[k*NC+n] ----------------
__global__ void k_transpose(const float* __restrict__ W, float* __restrict__ Wt,
                            int K, int NC) {
    int i = blockIdx.x * blockDim.x + threadIdx.x;
    if (i < K * NC) {
        int k = i / NC, n = i % NC;
        Wt[(size_t)n * K + k] = W[i];
    }
}

// ---------------- WMMA f32 GEMM: O = A[M,K] @ W[K,NC] (+bias, +leaky) ------
// One wave computes a 16 x (16*CPW) output strip via V_WMMA_F32_16X16X4_F32,
// reusing one A fragment across CPW column tiles (CPW accumulators).
// W is pre-transposed (Wt[NC,K]) so both A and B fragments are contiguous
// b64 loads: lanes 0-15 hold K={k,k+1}, lanes 16-31 hold K={k+2,k+3}.
template <int K, int NC, int CPW, bool BIAS, bool LEAKY>
__global__ void k_gemm_wmma(const float* __restrict__ A,
                            const float* __restrict__ Wt,   // [NC, K]
                            const float* __restrict__ bias,
                            float* __restrict__ O, int M) {
    const int lane  = threadIdx.x & 31;
    const int wave  = threadIdx.x >> 5;
    const int tile  = blockIdx.x * 4 + wave;          // 4 waves / block
    const int tilesN = NC / (16 * CPW);
    const int tilesTotal = (M / 16) * tilesN;
    if (tile >= tilesTotal) return;                   // wave-uniform guard
    const int row0 = (tile / tilesN) * 16;
    const int col0 = (tile % tilesN) * (16 * CPW);

    const int half = lane >> 4;                       // 0: K={0,1}, 1: K={2,3}
    const int l    = lane & 15;

    v8f acc[CPW];
#pragma unroll
    for (int j = 0; j < CPW; ++j) acc[j] = (v8f){};

    const float* ap = A  + (size_t)(row0 + l) * K + 2 * half;
    const float* bp = Wt + (size_t)(col0 + l) * K + 2 * half;

#pragma unroll 2
    for (int k = 0; k < K; k += 4) {
        v2f a = *reinterpret_cast<const v2f*>(ap);    // A[m][k+2h..k+2h+1]
#pragma unroll
        for (int j = 0; j < CPW; ++j) {
            v2f b = *reinterpret_cast<const v2f*>(bp + (size_t)j * 16 * K);
            acc[j] = __builtin_amdgcn_wmma_f32_16x16x4_f32(
                         /*neg_a=*/false, a, /*neg_b=*/false, b,
                         /*c_mod=*/(short)0, acc[j],
                         /*reuse_a=*/false, /*reuse_b=*/false);
        }
        ap += 4;
        bp += 4;
    }

#pragma unroll
    for (int j = 0; j < CPW; ++j) {
        const int cb = col0 + 16 * j + l;
        const float bb = BIAS ? bias[cb] : 0.0f;
#pragma unroll
        for (int r = 0; r < 8; ++r) {
            const int m = row0 + r + 8 * half;
            float v = acc[j][r] + bb;
            if (LEAKY) v = v > 0.0f ? v : NEG_SLOPE * v;
            O[(size_t)m * NC + cb] = v;
        }
    }
}

// ---------------- aggregation ----------------
// out[v][:] = t[v][:] * dinv[v]^2 + bias   (self-loop term + conv bias)
__global__ void k_agg_init(const float* __restrict__ t,
                           const float* __restrict__ dinv,
                           const float* __restrict__ bias,
                           float* __restrict__ out, int n) {
    int gid = blockIdx.x * blockDim.x + threadIdx.x;  // n*32 threads, 4 ch each
    int v = gid >> 5;
    if (v >= n) return;
    int c = (gid & 31) << 2;
    float di = dinv[v];
    float w = di * di;
    float4 tv = *reinterpret_cast<const float4*>(t + (size_t)v * 128 + c);
    float4 bv = *reinterpret_cast<const float4*>(bias + c);
    float4 o;
    o.x = tv.x * w + bv.x;
    o.y = tv.y * w + bv.y;
    o.z = tv.z * w + bv.z;
    o.w = tv.w * w + bv.w;
    *reinterpret_cast<float4*>(out + (size_t)v * 128 + c) = o;
}

// out[dst][:] += t[src][:] * dinv[src]*dinv[dst]   (one wave per edge,
// each lane owns 4 channels -> float4 gather + 4 global f32 atomics)
__global__ void k_agg_edges(const int* __restrict__ src,
                            const int* __restrict__ dst,
                            const float* __restrict__ dinv,
                            const float* __restrict__ t,
                            float* out, int E) {
    long long gid = (long long)blockIdx.x * blockDim.x + threadIdx.x;
    int e = (int)(gid >> 5);
    if (e >= E) return;
    int c = ((int)gid & 31) << 2;
    int s = src[e];
    int d = dst[e];
    float nrm = dinv[s] * dinv[d];
    float4 m = *reinterpret_cast<const float4*>(t + (size_t)s * 128 + c);
    float* o = out + (size_t)d * 128 + c;
    atomicAdd(o + 0, m.x * nrm);
    atomicAdd(o + 1, m.y * nrm);
    atomicAdd(o + 2, m.z * nrm);
    atomicAdd(o + 3, m.w * nrm);
}

__global__ void k_leaky(float* h, long long n) {
    long long i = (long long)blockIdx.x * blockDim.x + threadIdx.x;
    if (i < n) {
        float v = h[i];
        h[i] = v > 0.0f ? v : NEG_SLOPE * v;
    }
}

// ---------------- driver ----------------
extern "C" void kernel_launch(void* const* d_in, const int* in_sizes, int n_in,
                              void* d_out, int out_size, void* d_ws, size_t ws_size,
                              hipStream_t stream) {
    const float* x      = (const float*)d_in[0];   // [N, 256]
    const int*   ei     = (const int*)d_in[1];     // [2, E]
    const float* enc_W  = (const float*)d_in[2];   // [256, 128]
    const float* enc_b  = (const float*)d_in[3];   // [128]
    const float* conv_W = (const float*)d_in[4];   // [2, 128, 128]
    const float* conv_b = (const float*)d_in[5];   // [2, 128]
    const float* dec_W  = (const float*)d_in[6];   // [128, 64]
    const float* dec_b  = (const float*)d_in[7];   // [64]
    float* out = (float*)d_out;                    // [N, 64]

    const int N = in_sizes[0] / 256;               // 50000
    const int E = in_sizes[1] / 2;                 // 800000
    const int* e_src = ei;
    const int* e_dst = ei + E;

    // workspace layout
    float* dinv = (float*)d_ws;
    float* hbuf = dinv + ((N + 255) & ~255);       // [N,128]
    float* tbuf = hbuf + (size_t)N * 128;          // [N,128]
    float* wt   = tbuf + (size_t)N * 128;          // transposed weights (<=32K f32)

    // gcn_norm: dinv = rsqrt(deg(dst) + self-loop)
    k_deg_init <<<(N + 255) / 256, 256, 0, stream>>>(dinv, N);
    k_deg_accum<<<(E + 255) / 256, 256, 0, stream>>>(e_dst, dinv, E);
    k_deg_rsqrt<<<(N + 255) / 256, 256, 0, stream>>>(dinv, N);

    // wave-strip counts: each wave covers 16 rows x 64 cols (CPW=4)
    const int strips128 = (N / 16) * (128 / 64);   // 6250
    const int strips64  = (N / 16) * (64 / 64);    // 3125

    // encoder: h = leaky_relu(x @ enc_W + enc_b)
    k_transpose<<<(256 * 128 + 255) / 256, 256, 0, stream>>>(enc_W, wt, 256, 128);
    k_gemm_wmma<256, 128, 4, true, true>
        <<<(strips128 + 3) / 4, 128, 0, stream>>>(x, wt, enc_b, hbuf, N);

    for (int layer = 0; layer < 2; ++layer) {
        // t = h @ conv_W[layer]
        k_transpose<<<(128 * 128 + 255) / 256, 256, 0, stream>>>(
            conv_W + (size_t)layer * 128 * 128, wt, 128, 128);
        k_gemm_wmma<128, 128, 4, false, false>
            <<<(strips128 + 3) / 4, 128, 0, stream>>>(hbuf, wt, nullptr, tbuf, N);
        // h = t*dinv^2 + b  (self-loop) ...
        k_agg_init<<<((long long)N * 32 + 255) / 256, 256, 0, stream>>>(
            tbuf, dinv, conv_b + (size_t)layer * 128, hbuf, N);
        // ... += scatter of normalized messages
        k_agg_edges<<<(int)(((long long)E * 32 + 255) / 256), 256, 0, stream>>>(
            e_src, e_dst, dinv, tbuf, hbuf, E);
        // h = leaky_relu(h)
        k_leaky<<<(int)(((long long)N * 128 + 255) / 256), 256, 0, stream>>>(
            hbuf, (long long)N * 128);
    }

    // decoder: out = h @ dec_W + dec_b
    k_transpose<<<(128 * 64 + 255) / 256, 256, 0, stream>>>(dec_W, wt, 128, 64);
    k_gemm_wmma<128, 64, 4, true, false>
        <<<(strips64 + 3) / 4, 128, 0, stream>>>(hbuf, wt, dec_b, out, N);
}